// GCNConv_37056977830252
// MI455X (gfx1250) — compile-verified
//
#include <hip/hip_runtime.h>
#include <cstdint>
#include <cstddef>

#define IN_FEATS   256
#define OUT_FEATS  128
#define TILE_M     16
#define LDS_STRIDE 260   // 256 + 4 pad: A-frag reads spread across all 64 LDS banks

typedef __attribute__((ext_vector_type(2))) float v2f;
typedef __attribute__((ext_vector_type(8))) float v8f;

// ---------------------------------------------------------------------------
// Zero fill (grid-stride)
// ---------------------------------------------------------------------------
__global__ __launch_bounds__(256) void zero_f32_kernel(float* __restrict__ p, long n) {
  long i = (long)blockIdx.x * blockDim.x + threadIdx.x;
  const long stride = (long)gridDim.x * blockDim.x;
  for (; i < n; i += stride) p[i] = 0.0f;
}

// ---------------------------------------------------------------------------
// Degree accumulation: deg_out[src[e]] += 1, deg_in[dst[e]] += 1
// ---------------------------------------------------------------------------
__global__ __launch_bounds__(256) void degree_kernel(const int* __restrict__ src,
                                                     const int* __restrict__ dst,
                                                     float* __restrict__ deg_out,
                                                     float* __restrict__ deg_in,
                                                     int n_edges) {
  const int i = blockIdx.x * blockDim.x + threadIdx.x;
  if (i < n_edges) {
    unsafeAtomicAdd(&deg_out[src[i]], 1.0f);   // hw global_atomic_add_f32
    unsafeAtomicAdd(&deg_in[dst[i]],  1.0f);
  }
}

// ---------------------------------------------------------------------------
// h = (feat * rsqrt(max(deg_out,1))) @ weight   via V_WMMA_F32_16X16X4_F32
//   Covers FULL 16-row tiles only (no bounds checks anywhere in the kernel).
//   block: 256 threads = 8 waves; wave w owns N-tile [16w, 16w+16)
//   block m: rows [16m, 16m+16); K = 256 stepped by 4
// ---------------------------------------------------------------------------
__global__ __launch_bounds__(256) void gemm_wmma_kernel(const float* __restrict__ feat,
                                                        const float* __restrict__ deg_out,
                                                        const float* __restrict__ weight,
                                                        float* __restrict__ h) {
  __shared__ float tileA[TILE_M * LDS_STRIDE];

  const int m0 = blockIdx.x * TILE_M;
  const int t  = threadIdx.x;              // 0..255 == column of the A tile

  // Stage pre-scaled 16x256 A tile into LDS (fully coalesced: 256 lanes = 1 row)
  #pragma unroll
  for (int r = 0; r < TILE_M; ++r) {
    const int row = m0 + r;
    const float s = rsqrtf(fmaxf(deg_out[row], 1.0f));
    tileA[r * LDS_STRIDE + t] = feat[(size_t)row * IN_FEATS + t] * s;
  }
  __syncthreads();

  const int wave = t >> 5;
  const int lane = t & 31;
  const int half = lane >> 4;              // 0: lanes 0-15, 1: lanes 16-31
  const int l    = lane & 15;
  const int n0   = wave * 16;

  // A frag (16x4 f32): lane l row=l; low half K=k..k+1, high half K=k+2..k+3
  const float* Abase = &tileA[l * LDS_STRIDE + 2 * half];
  // B frag (4x16 f32): lane l col=n0+l; low half K=k..k+1, high half K=k+2..k+3
  const float* Wbase = weight + n0 + l;

  v8f c = {0.f, 0.f, 0.f, 0.f, 0.f, 0.f, 0.f, 0.f};

  #pragma unroll 8
  for (int k = 0; k < IN_FEATS; k += 4) {
    v2f a = *(const v2f*)(Abase + k);                          // ds_load_2addr_b64
    v2f b;
    b.x = Wbase[(size_t)(k + 2 * half    ) * OUT_FEATS];
    b.y = Wbase[(size_t)(k + 2 * half + 1) * OUT_FEATS];
    c = __builtin_amdgcn_wmma_f32_16x16x4_f32(
            /*neg_a=*/false, a, /*neg_b=*/false, b,
            /*c_mod=*/(short)0, c, /*reuse_a=*/false, /*reuse_b=*/false);
  }

  // D layout: VGPR v -> M = v (lanes 0-15) / v+8 (lanes 16-31), N = n0 + l
  // Straight-line, guard-free: 8 coalesced dword stores.
  float* hout = h + (size_t)(m0 + 8 * half) * OUT_FEATS + n0 + l;
  #pragma unroll
  for (int v = 0; v < 8; ++v) {
    hout[(size_t)v * OUT_FEATS] = c[v];
  }
}

// ---------------------------------------------------------------------------
// Remainder rows (n_nodes % 16) — scalar fallback; never launched for N=100000
// ---------------------------------------------------------------------------
__global__ __launch_bounds__(128) void gemm_tail_kernel(const float* __restrict__ feat,
                                                        const float* __restrict__ deg_out,
                                                        const float* __restrict__ weight,
                                                        float* __restrict__ h,
                                                        int row_base) {
  const int row = row_base + blockIdx.x;
  const int col = threadIdx.x;             // 0..127
  const float s = rsqrtf(fmaxf(deg_out[row], 1.0f));
  float acc = 0.0f;
  for (int k = 0; k < IN_FEATS; ++k)
    acc += feat[(size_t)row * IN_FEATS + k] * s * weight[(size_t)k * OUT_FEATS + col];
  h[(size_t)row * OUT_FEATS + col] = acc;
}

// ---------------------------------------------------------------------------
// Edge scatter: out[dst[e]] += h[src[e]] * ew[e]
//   one wave per edge: lane l handles cols 4l..4l+3 (float4 gather = 512B/wave)
// ---------------------------------------------------------------------------
__global__ __launch_bounds__(256) void scatter_kernel(const float* __restrict__ h,
                                                      const int* __restrict__ src,
                                                      const int* __restrict__ dst,
                                                      const float* __restrict__ ew,
                                                      float* __restrict__ out,
                                                      int n_edges) {
  const int e    = (int)(((long)blockIdx.x * blockDim.x + threadIdx.x) >> 5);
  const int lane = threadIdx.x & 31;
  if (e >= n_edges) return;

  const int   s = src[e];
  const int   d = dst[e];
  const float w = ew[e];

  // Prefetch the gather row a few edges ahead (global_prefetch_b8)
  if (e + 8 < n_edges) {
    __builtin_prefetch(h + (size_t)src[e + 8] * OUT_FEATS + 4 * lane, 0, 3);
  }

  const float4 v = ((const float4*)(h + (size_t)s * OUT_FEATS))[lane];
  float* o = out + (size_t)d * OUT_FEATS + 4 * lane;
  unsafeAtomicAdd(o + 0, v.x * w);
  unsafeAtomicAdd(o + 1, v.y * w);
  unsafeAtomicAdd(o + 2, v.z * w);
  unsafeAtomicAdd(o + 3, v.w * w);
}

// ---------------------------------------------------------------------------
// Finalize: out = out * rsqrt(max(deg_in,1)) + bias   (float4 vectorized)
// ---------------------------------------------------------------------------
__global__ __launch_bounds__(256) void finalize_kernel(float4* __restrict__ out,
                                                       const float* __restrict__ deg_in,
                                                       const float4* __restrict__ bias4,
                                                       long n_vec4) {
  const long i = (long)blockIdx.x * blockDim.x + threadIdx.x;
  if (i >= n_vec4) return;
  const int row  = (int)(i >> 5);                      // 32 float4 per row
  const int cg   = (int)(i & 31);                      // col group
  const float s  = rsqrtf(fmaxf(deg_in[row], 1.0f));
  const float4 b = bias4[cg];
  float4 v = out[i];
  v.x = v.x * s + b.x;
  v.y = v.y * s + b.y;
  v.z = v.z * s + b.z;
  v.w = v.w * s + b.w;
  out[i] = v;
}

// ---------------------------------------------------------------------------
// Host launcher
// inputs: feat[N*256] f32, src[E] i32, dst[E] i32, ew[E] f32,
//         weight[256*128] f32, bias[128] f32; out: [N*128] f32
// ---------------------------------------------------------------------------
extern "C" void kernel_launch(void* const* d_in, const int* in_sizes, int n_in,
                              void* d_out, int out_size, void* d_ws, size_t ws_size,
                              hipStream_t stream) {
  const float* feat   = (const float*)d_in[0];
  const int*   src    = (const int*)  d_in[1];
  const int*   dst    = (const int*)  d_in[2];
  const float* ew     = (const float*)d_in[3];
  const float* weight = (const float*)d_in[4];
  const float* bias   = (const float*)d_in[5];
  float*       out    = (float*)d_out;

  const int n_nodes = in_sizes[0] / IN_FEATS;
  const int n_edges = in_sizes[1];

  // workspace layout: h [N*128] | deg_out [N] | deg_in [N]
  float* h       = (float*)d_ws;
  float* deg_out = h + (size_t)n_nodes * OUT_FEATS;
  float* deg_in  = deg_out + n_nodes;

  // 1) zero degree vectors (contiguous 2N) and output accumulator
  {
    const long n = 2L * n_nodes;
    zero_f32_kernel<<<(int)((n + 255) / 256), 256, 0, stream>>>(deg_out, n);
  }
  {
    const long n = (long)n_nodes * OUT_FEATS;
    zero_f32_kernel<<<(int)((n + 255) / 256), 256, 0, stream>>>(out, n);
  }

  // 2) degrees
  degree_kernel<<<(n_edges + 255) / 256, 256, 0, stream>>>(src, dst, deg_out, deg_in, n_edges);

  // 3) projection GEMM on the fp32 WMMA pipe (full tiles; tail kernel for rest)
  {
    const int full_tiles = n_nodes / TILE_M;
    const int rem        = n_nodes - full_tiles * TILE_M;
    if (full_tiles > 0)
      gemm_wmma_kernel<<<full_tiles, 256, 0, stream>>>(feat, deg_out, weight, h);
    if (rem > 0)   // never taken for N=100000 (6250 full tiles)
      gemm_tail_kernel<<<rem, OUT_FEATS, 0, stream>>>(feat, deg_out, weight, h,
                                                      full_tiles * TILE_M);
  }

  // 4) edge scatter-add (8 edges per 256-thread block, one wave each)
  {
    const long blocks = ((long)n_edges * 32 + 255) / 256;
    scatter_kernel<<<(int)blocks, 256, 0, stream>>>(h, src, dst, ew, out, n_edges);
  }

  // 5) post-scale + bias (float4 wide)
  {
    const long n4 = (long)n_nodes * (OUT_FEATS / 4);
    finalize_kernel<<<(int)((n4 + 255) / 256), 256, 0, stream>>>(
        (float4*)out, deg_in, (const float4*)bias, n4);
  }
}